// AdaptivePiecewiseConv2d_60361470378304
// MI455X (gfx1250) — compile-verified
//
#include <hip/hip_runtime.h>

// AdaptivePiecewiseConv2d for MI455X (gfx1250, wave32).
// y[n,o] = (X·A0)[n,o] + Σ_i b0[i,o] + Σ_i [x≥p1]·(da·x+db)
// Base GEMM -> V_WMMA_F32_16X16X4_F32 (exact f32); correction -> VALU.

typedef float v2f __attribute__((ext_vector_type(2)));
typedef float v8f __attribute__((ext_vector_type(8)));

#define IN_CH      16
#define OUT_CH     32
#define NUM_INPUTS 144           // IN_CH * 3 * 3
#define NPIX       1024          // 32*32
#define NTOT       8192          // BATCH * NPIX
#define LDSSTRIDE  145           // 144 padded: odd dword stride -> bank-conflict-free

// ---------------------------------------------------------------------------
// Prep: per (i,o) pair, convert 3 breakpoints into segment slope/intercepts.
// Tables laid out [i][o] (i.e. t = i*32 + o) so WMMA B loads are coalesced.
// ---------------------------------------------------------------------------
__global__ void apw_prep_coeffs(const float* __restrict__ pos,
                                const float* __restrict__ val,
                                float* __restrict__ a0_t,
                                float* __restrict__ b0_t,
                                float* __restrict__ p1_t,
                                float* __restrict__ da_t,
                                float* __restrict__ db_t) {
    int t = blockIdx.x * blockDim.x + threadIdx.x;
    if (t >= NUM_INPUTS * OUT_CH) return;
    const float* p = pos + t * 3;
    const float* v = val + t * 3;
    float p0 = p[0], p1 = p[1], p2 = p[2];
    float v0 = v[0], v1 = v[1], v2 = v[2];
    float a0 = (v1 - v0) / (p1 - p0);
    float b0 = fmaf(-p0, a0, v0);
    float a1 = (v2 - v1) / (p2 - p1);
    float b1 = fmaf(-p1, a1, v1);
    a0_t[t] = a0;
    b0_t[t] = b0;
    p1_t[t] = p1;
    da_t[t] = a1 - a0;
    db_t[t] = b1 - b0;
}

// ---------------------------------------------------------------------------
// Main: one block = 16 patch rows. Two waves: wave w handles output cols
// [16w, 16w+16). Patch tile staged once in LDS (16 x 144, stride 145 dwords).
// WMMA base GEMM over K=144 in 36 steps of 16x16x4, then VALU piecewise
// correction accumulated into the same C fragment.
// ---------------------------------------------------------------------------
__global__ __launch_bounds__(64)
void apw_conv_main(const float* __restrict__ x,
                   const float* __restrict__ a0_t,
                   const float* __restrict__ b0_t,
                   const float* __restrict__ p1_t,
                   const float* __restrict__ da_t,
                   const float* __restrict__ db_t,
                   float* __restrict__ out) {
    __shared__ float tile[16 * LDSSTRIDE];

    const int n0  = blockIdx.x << 4;   // first of 16 patch rows
    const int tid = threadIdx.x;

    // ---- stage im2col tile: 2304 elements, 36 full-wave iterations ----
    for (int idx = tid; idx < 16 * NUM_INPUTS; idx += 64) {
        int row = idx / NUM_INPUTS;
        int i   = idx - row * NUM_INPUTS;
        int n   = n0 + row;
        int b   = n >> 10;
        int hw  = n & (NPIX - 1);
        int ho  = hw >> 5;
        int wo  = hw & 31;
        int c   = i / 9;
        int kk  = i - c * 9;
        int kh  = kk / 3;
        int kw  = kk - kh * 3;
        int yi  = ho + kh - 1;
        int xi  = wo + kw - 1;
        float v = 0.0f;
        if ((unsigned)yi < 32u && (unsigned)xi < 32u)
            v = x[((b * IN_CH + c) * 32 + yi) * 32 + xi];
        tile[row * LDSSTRIDE + i] = v;
    }
    __syncthreads();

    const int lane  = tid & 31;
    const int otile = tid >> 5;        // 0 or 1 (which 16-wide O tile)
    const int hi    = lane >> 4;       // half-wave selector
    const int col   = lane & 15;
    const int N     = otile * 16 + col;

    // ---- base GEMM via f32 WMMA: C += A(16x4) * B(4x16) over 36 K-steps ----
    // A frag: lane M = col, K = v + 2*hi. B frag: lane N = col, K = v + 2*hi.
    v8f acc = {};
    for (int k0 = 0; k0 < NUM_INPUTS; k0 += 4) {
        v2f a, b;
        a.x = tile[col * LDSSTRIDE + k0 + 0 + 2 * hi];
        a.y = tile[col * LDSSTRIDE + k0 + 1 + 2 * hi];
        b.x = a0_t[(k0 + 0 + 2 * hi) * OUT_CH + N];
        b.y = a0_t[(k0 + 1 + 2 * hi) * OUT_CH + N];
        acc = __builtin_amdgcn_wmma_f32_16x16x4_f32(
            /*neg_a=*/false, a, /*neg_b=*/false, b,
            /*c_mod=*/(short)0, acc, /*reuse_a=*/false, /*reuse_b=*/false);
    }

    // ---- piecewise correction + intercept sum (matches C-fragment layout) ----
    float corr[8] = {0.f, 0.f, 0.f, 0.f, 0.f, 0.f, 0.f, 0.f};
    float bacc = 0.0f;
    for (int i = 0; i < NUM_INPUTS; ++i) {
        float p1v = p1_t[i * OUT_CH + N];
        float dav = da_t[i * OUT_CH + N];
        float dbv = db_t[i * OUT_CH + N];
        bacc += b0_t[i * OUT_CH + N];
#pragma unroll
        for (int r = 0; r < 8; ++r) {
            // element (M = r + 8*hi, N): broadcast LDS read per half-wave
            float xv = tile[(r + 8 * hi) * LDSSTRIDE + i];
            corr[r] += (xv >= p1v) ? fmaf(dav, xv, dbv) : 0.0f;
        }
    }

    // ---- store: out[b, o, ho, wo] = b*32768 + o*1024 + hw ----
#pragma unroll
    for (int r = 0; r < 8; ++r) {
        int n  = n0 + r + 8 * hi;
        int b  = n >> 10;
        int hw = n & (NPIX - 1);
        out[(b * OUT_CH + N) * NPIX + hw] = acc[r] + corr[r] + bacc;
    }
}

// ---------------------------------------------------------------------------
extern "C" void kernel_launch(void* const* d_in, const int* in_sizes, int n_in,
                              void* d_out, int out_size, void* d_ws, size_t ws_size,
                              hipStream_t stream) {
    const float* x   = (const float*)d_in[0];   // (8,16,32,32)
    const float* pos = (const float*)d_in[1];   // (144,32,3)
    const float* val = (const float*)d_in[2];   // (144,32,3)
    float* out = (float*)d_out;                 // (8,32,32,32)

    const int pairs = NUM_INPUTS * OUT_CH;      // 4608
    float* ws   = (float*)d_ws;                 // 5 tables * 18.4 KB = 92 KB
    float* a0_t = ws;
    float* b0_t = ws + 1 * pairs;
    float* p1_t = ws + 2 * pairs;
    float* da_t = ws + 3 * pairs;
    float* db_t = ws + 4 * pairs;

    apw_prep_coeffs<<<(pairs + 255) / 256, 256, 0, stream>>>(
        pos, val, a0_t, b0_t, p1_t, da_t, db_t);

    apw_conv_main<<<NTOT / 16, 64, 0, stream>>>(
        x, a0_t, b0_t, p1_t, da_t, db_t, out);
}